// GAT_44796508897551
// MI455X (gfx1250) — compile-verified
//
#include <hip/hip_runtime.h>
#include <math.h>

#define NNODES 50000
#define K1PAD  168   // 165 rounded up to multiple of 8

typedef float v2f __attribute__((ext_vector_type(2)));
typedef float v8f __attribute__((ext_vector_type(8)));

// ---------------- device helpers ----------------

__device__ __forceinline__ void atomicMaxF(float* addr, float v) {
  // float ordering == int ordering for v>=0, reversed unsigned ordering for v<0
  if (v >= 0.0f) atomicMax((int*)addr, __float_as_int(v));
  else           atomicMin((unsigned int*)addr, __float_as_uint(v));
}

__device__ __forceinline__ void edge_nodes(const int* __restrict__ ei, int e, int E,
                                           int& s, int& d) {
  if (e < E) { s = ei[e]; d = ei[E + e]; }    // edge_index is [2,E]
  else       { s = d = e - E; }               // implicit self-loops
}

// ---------------- utility kernels ----------------

__global__ __launch_bounds__(256) void k_fill(float* __restrict__ p, float v, int n) {
  int i = blockIdx.x * blockDim.x + threadIdx.x;
  if (i < n) p[i] = v;
}

// dst[r*Kdst + c] = (c < Ksrc) ? src[r*Ksrc + c] : 0   (pad inner dim with zeros)
__global__ __launch_bounds__(256) void k_pad_rows(const float* __restrict__ src,
                                                  float* __restrict__ dst,
                                                  int Nr, int Ksrc, int Kdst) {
  int i = blockIdx.x * blockDim.x + threadIdx.x;
  if (i >= Nr * Kdst) return;
  int r = i / Kdst, c = i - r * Kdst;
  dst[i] = (c < Ksrc) ? src[(size_t)r * Ksrc + c] : 0.f;
}

// dst[i] = (i < nsrc) ? src[i] : 0  (copy with zero tail; pads trailing rows)
__global__ __launch_bounds__(256) void k_copy_ztail(const float* __restrict__ src,
                                                    float* __restrict__ dst,
                                                    int nsrc, int ntotal) {
  int i = blockIdx.x * blockDim.x + threadIdx.x;
  if (i >= ntotal) return;
  dst[i] = (i < nsrc) ? src[i] : 0.f;
}

// ---------------- WMMA GEMM ----------------
// C[M,Ncol] = A[M,K] @ B[K,Ncol], fp32 WMMA 16x16x4.
// One wave computes a 16 x (16*COLT) strip. Per K-step: batch-load the A
// fragment plus all COLT B fragments (single clause, one wait), then run COLT
// independent back-to-back WMMAs. Pointer increments only (no addr mul in loop).
// K must be a multiple of 4.
template<int COLT>
__global__ __launch_bounds__(256) void k_wmma_gemm(const float* __restrict__ A,
                                                   const float* __restrict__ B,
                                                   float* __restrict__ C,
                                                   int M, int K, int Ncol) {
  const int wave = (blockIdx.x * blockDim.x + threadIdx.x) >> 5;
  const int lane = threadIdx.x & 31;
  const int groupsN = Ncol / (16 * COLT);
  const int rowTile = wave / groupsN;
  const int grp     = wave % groupsN;
  if (rowTile * 16 >= M) return;              // wave-uniform -> EXEC all-ones below
  const bool hi  = lane >= 16;
  const int  l16 = lane & 15;
  const int  arow  = rowTile * 16 + l16;      // A: lanes 0-15 K{0,1}, 16-31 K{2,3}
  const int  bcol0 = grp * 16 * COLT + l16;   // B: col n in lane n%16

  v8f acc[COLT];
#pragma unroll
  for (int t = 0; t < COLT; ++t) acc[t] = (v8f){};

  const float* aptr = A + (size_t)arow * K + (hi ? 2 : 0);
  const float* bptr = B + (size_t)(hi ? 2 : 0) * Ncol + bcol0;
  const size_t bstep = (size_t)4 * Ncol;

  for (int k0 = 0; k0 < K; k0 += 4) {
    // ---- batch all loads first: one clause, one wait ----
    v2f a;
    a.x = aptr[0];                            // adjacent -> global_load_b64
    a.y = aptr[1];
    v2f b[COLT];
#pragma unroll
    for (int t = 0; t < COLT; ++t) {
      b[t].x = bptr[t * 16];                  // const immediate offsets
      b[t].y = bptr[(size_t)Ncol + t * 16];
    }
    // ---- COLT independent WMMAs back-to-back ----
#pragma unroll
    for (int t = 0; t < COLT; ++t)
      acc[t] = __builtin_amdgcn_wmma_f32_16x16x4_f32(false, a, false, b[t],
                                                     (short)0, acc[t], false, false);
    aptr += 4;
    bptr += bstep;
  }

  // C/D layout: VGPR v, lanes 0-15 -> row v, lanes 16-31 -> row v+8
  const int orow = rowTile * 16 + (hi ? 8 : 0);
  float* cptr = C + (size_t)orow * Ncol + bcol0;
#pragma unroll
  for (int t = 0; t < COLT; ++t)
#pragma unroll
    for (int v = 0; v < 8; ++v)
      cptr[(size_t)v * Ncol + t * 16] = acc[t][v];
}

// ---------------- attention coefficients ----------------
// es[n,h] = <h[n,h,:], a_src[h,:]>, ed likewise. One wave per (node, head).
__global__ __launch_bounds__(256) void k_attn(const float* __restrict__ h,
                                              const float* __restrict__ as_,
                                              const float* __restrict__ ad_,
                                              float* __restrict__ es,
                                              float* __restrict__ ed,
                                              int Nn, int H, int C) {
  const int wave = (blockIdx.x * blockDim.x + threadIdx.x) >> 5;
  const int lane = threadIdx.x & 31;
  if (wave >= Nn * H) return;
  const int n = wave / H, hd = wave % H;
  float ss = 0.f, dd = 0.f;
  for (int c = lane; c < C; c += 32) {
    float v = h[(size_t)n * H * C + hd * C + c];
    ss += v * as_[hd * C + c];
    dd += v * ad_[hd * C + c];
  }
  for (int off = 16; off > 0; off >>= 1) {
    ss += __shfl_down(ss, off);
    dd += __shfl_down(dd, off);
  }
  if (lane == 0) { es[n * H + hd] = ss; ed[n * H + hd] = dd; }
}

// ---------------- edge softmax passes ----------------

__global__ __launch_bounds__(256) void k_edge_max(const int* __restrict__ ei,
                                                  const float* __restrict__ es,
                                                  const float* __restrict__ ed,
                                                  float* __restrict__ m,
                                                  int E, int Nn, int H) {
  int i = blockIdx.x * blockDim.x + threadIdx.x;
  if (i >= (E + Nn) * H) return;
  int e = i / H, hd = i - e * H;
  int s, d; edge_nodes(ei, e, E, s, d);
  float v = es[s * H + hd] + ed[d * H + hd];
  v = v > 0.f ? v : 0.2f * v;                 // leaky_relu
  atomicMaxF(&m[d * H + hd], v);
}

__global__ __launch_bounds__(256) void k_edge_expsum(const int* __restrict__ ei,
                                                     const float* __restrict__ es,
                                                     const float* __restrict__ ed,
                                                     const float* __restrict__ m,
                                                     float* __restrict__ ssum,
                                                     int E, int Nn, int H) {
  int i = blockIdx.x * blockDim.x + threadIdx.x;
  if (i >= (E + Nn) * H) return;
  int e = i / H, hd = i - e * H;
  int s, d; edge_nodes(ei, e, E, s, d);
  float v = es[s * H + hd] + ed[d * H + hd];
  v = v > 0.f ? v : 0.2f * v;
  unsafeAtomicAdd(&ssum[d * H + hd], __expf(v - m[d * H + hd]));
}

// out[dst,h,:] += alpha * h[src,h,:]. One wave per (edge, head).
// C % 4 == 0 path: one float4 (b128) gather per lane + 4 f32 atomics.
template<int C>
__global__ __launch_bounds__(256) void k_edge_agg(const int* __restrict__ ei,
                                                  const float* __restrict__ es,
                                                  const float* __restrict__ ed,
                                                  const float* __restrict__ m,
                                                  const float* __restrict__ ssum,
                                                  const float* __restrict__ h,
                                                  float* __restrict__ out,
                                                  int E, int Nn, int H) {
  const int wave = (blockIdx.x * blockDim.x + threadIdx.x) >> 5;
  const int lane = threadIdx.x & 31;
  if (wave >= (E + Nn) * H) return;
  int e = wave / H, hd = wave - e * H;
  int s, d; edge_nodes(ei, e, E, s, d);
  float v = es[s * H + hd] + ed[d * H + hd];
  v = v > 0.f ? v : 0.2f * v;
  const float alpha = __expf(v - m[d * H + hd]) / (ssum[d * H + hd] + 1e-16f);
  const size_t srcBase = (size_t)s * H * C + (size_t)hd * C;
  const size_t dstBase = (size_t)d * H * C + (size_t)hd * C;
  if constexpr (C % 4 == 0 && C <= 128) {
    const int c = lane * 4;
    if (c < C) {                              // provably true for C==128
      const float4 hv = *reinterpret_cast<const float4*>(h + srcBase + c);
      unsafeAtomicAdd(&out[dstBase + c + 0], alpha * hv.x);
      unsafeAtomicAdd(&out[dstBase + c + 1], alpha * hv.y);
      unsafeAtomicAdd(&out[dstBase + c + 2], alpha * hv.z);
      unsafeAtomicAdd(&out[dstBase + c + 3], alpha * hv.w);
    }
  } else {
    for (int c = lane; c < C; c += 32)
      unsafeAtomicAdd(&out[dstBase + c], alpha * h[srcBase + c]);
  }
}

__global__ __launch_bounds__(256) void k_bias_act(float* __restrict__ x,
                                                  const float* __restrict__ b,
                                                  int Nn, int F, int doRelu) {
  int i = blockIdx.x * blockDim.x + threadIdx.x;
  if (i >= Nn * F) return;
  float v = x[i] + b[i % F];
  x[i] = doRelu ? fmaxf(v, 0.f) : v;
}

// Tiny GEMM for layer 3 (Ncol=2): scalar, one thread per output element.
__global__ __launch_bounds__(256) void k_gemm_small(const float* __restrict__ A,
                                                    const float* __restrict__ W,
                                                    float* __restrict__ out,
                                                    int Nn, int K, int F) {
  int i = blockIdx.x * blockDim.x + threadIdx.x;
  if (i >= Nn * F) return;
  int n = i / F, f = i - n * F;
  float acc = 0.f;
  for (int k = 0; k < K; ++k) acc += A[(size_t)n * K + k] * W[k * F + f];
  out[i] = acc;
}

// ---------------- host launcher ----------------

static inline int cdiv(long long a, int b) { return (int)((a + b - 1) / b); }

extern "C" void kernel_launch(void* const* d_in, const int* in_sizes, int n_in,
                              void* d_out, int out_size, void* d_ws, size_t ws_size,
                              hipStream_t stream) {
  const int N = NNODES;
  const int E = in_sizes[1] / 2;

  const float* x   = (const float*)d_in[0];
  const int*   ei  = (const int*)  d_in[1];
  const float* W1  = (const float*)d_in[2];
  const float* a1s = (const float*)d_in[3];
  const float* a1d = (const float*)d_in[4];
  const float* b1  = (const float*)d_in[5];
  const float* W2  = (const float*)d_in[6];
  const float* a2s = (const float*)d_in[7];
  const float* a2d = (const float*)d_in[8];
  const float* b2  = (const float*)d_in[9];
  const float* W3  = (const float*)d_in[10];
  const float* a3s = (const float*)d_in[11];
  const float* a3d = (const float*)d_in[12];
  const float* b3  = (const float*)d_in[13];
  float* out = (float*)d_out;

  float* bufA  = (float*)d_ws;                  // N*256 : linear output h
  float* bufB  = bufA  + (size_t)N * 256;       // N*256 : aggregated output / next x
  float* h3    = bufB  + (size_t)N * 256;       // N*2
  float* es    = h3    + (size_t)N * 2;         // N*2
  float* ed    = es    + (size_t)N * 2;         // N*2
  float* mb    = ed    + (size_t)N * 2;         // N*2   segment max
  float* sb    = mb    + (size_t)N * 2;         // N*2   segment expsum
  float* xpad  = sb    + (size_t)N * 2;         // N*K1PAD  (x zero-padded to K=168)
  float* W1pad = xpad  + (size_t)N * K1PAD;     // K1PAD*256

  const int TB = 256;
  const float NEGINF = -INFINITY;
  const int gemmWaves = ((N + 15) / 16) * (256 / 64);   // 3125 row tiles * 4 strip groups
  const int eh2 = (E + N) * 2;
  const int eh1 = (E + N);

  // ---------- pad inputs for guard-free WMMA loop (K=165 -> 168) ----------
  k_pad_rows<<<cdiv((long long)N * K1PAD, TB), TB, 0, stream>>>(x, xpad, N, 165, K1PAD);
  k_copy_ztail<<<cdiv(K1PAD * 256, TB), TB, 0, stream>>>(W1, W1pad, 165 * 256, K1PAD * 256);

  // ---------- Layer 1: 165 -> [2 heads x 128], concat, relu ----------
  k_wmma_gemm<4><<<cdiv((long long)gemmWaves * 32, TB), TB, 0, stream>>>(xpad, W1pad, bufA, N, K1PAD, 256);
  k_attn<<<cdiv((long long)N * 2 * 32, TB), TB, 0, stream>>>(bufA, a1s, a1d, es, ed, N, 2, 128);
  k_fill<<<cdiv(N * 2, TB), TB, 0, stream>>>(mb, NEGINF, N * 2);
  k_fill<<<cdiv(N * 2, TB), TB, 0, stream>>>(sb, 0.f, N * 2);
  k_fill<<<cdiv((long long)N * 256, TB), TB, 0, stream>>>(bufB, 0.f, N * 256);
  k_edge_max   <<<cdiv(eh2, TB), TB, 0, stream>>>(ei, es, ed, mb, E, N, 2);
  k_edge_expsum<<<cdiv(eh2, TB), TB, 0, stream>>>(ei, es, ed, mb, sb, E, N, 2);
  k_edge_agg<128><<<cdiv((long long)eh2 * 32, TB), TB, 0, stream>>>(ei, es, ed, mb, sb, bufA, bufB, E, N, 2);
  k_bias_act<<<cdiv((long long)N * 256, TB), TB, 0, stream>>>(bufB, b1, N, 256, 1);

  // ---------- Layer 2: 256 -> [2 heads x 128], concat, relu ----------
  k_wmma_gemm<4><<<cdiv((long long)gemmWaves * 32, TB), TB, 0, stream>>>(bufB, W2, bufA, N, 256, 256);
  k_attn<<<cdiv((long long)N * 2 * 32, TB), TB, 0, stream>>>(bufA, a2s, a2d, es, ed, N, 2, 128);
  k_fill<<<cdiv(N * 2, TB), TB, 0, stream>>>(mb, NEGINF, N * 2);
  k_fill<<<cdiv(N * 2, TB), TB, 0, stream>>>(sb, 0.f, N * 2);
  k_fill<<<cdiv((long long)N * 256, TB), TB, 0, stream>>>(bufB, 0.f, N * 256);
  k_edge_max   <<<cdiv(eh2, TB), TB, 0, stream>>>(ei, es, ed, mb, E, N, 2);
  k_edge_expsum<<<cdiv(eh2, TB), TB, 0, stream>>>(ei, es, ed, mb, sb, E, N, 2);
  k_edge_agg<128><<<cdiv((long long)eh2 * 32, TB), TB, 0, stream>>>(ei, es, ed, mb, sb, bufA, bufB, E, N, 2);
  k_bias_act<<<cdiv((long long)N * 256, TB), TB, 0, stream>>>(bufB, b2, N, 256, 1);

  // ---------- Layer 3: 256 -> [1 head x 2], mean(=identity), no relu ----------
  k_gemm_small<<<cdiv(N * 2, TB), TB, 0, stream>>>(bufB, W3, h3, N, 256, 2);
  k_attn<<<cdiv((long long)N * 32, TB), TB, 0, stream>>>(h3, a3s, a3d, es, ed, N, 1, 2);
  k_fill<<<cdiv(N, TB), TB, 0, stream>>>(mb, NEGINF, N);
  k_fill<<<cdiv(N, TB), TB, 0, stream>>>(sb, 0.f, N);
  k_fill<<<cdiv(N * 2, TB), TB, 0, stream>>>(out, 0.f, N * 2);
  k_edge_max   <<<cdiv(eh1, TB), TB, 0, stream>>>(ei, es, ed, mb, E, N, 1);
  k_edge_expsum<<<cdiv(eh1, TB), TB, 0, stream>>>(ei, es, ed, mb, sb, E, N, 1);
  k_edge_agg<2><<<cdiv((long long)eh1 * 32, TB), TB, 0, stream>>>(ei, es, ed, mb, sb, h3, out, E, N, 1);
  k_bias_act<<<cdiv(N * 2, TB), TB, 0, stream>>>(out, b3, N, 2, 0);
}